// TransformerWord2VecEncoder_37804302139707
// MI455X (gfx1250) — compile-verified
//
#include <hip/hip_runtime.h>
#include <hip/hip_bf16.h>
#include <stdint.h>

// Problem geometry (matches reference)
#define BATCH   256
#define SEQ     512
#define VOCAB   100000
#define DIM     64                  // floats per embedding row (256 bytes)
#define CHUNKS  16                  // 16-byte chunks per row
#define THREADS 256                 // 8 waves (wave32)
#define ITERS   4                   // rows-in-flight multiplier per thread group
#define ROWS_PER_BLOCK ((THREADS / CHUNKS) * ITERS)   // 64 rows / block
#define NPOS    (BATCH * SEQ)

typedef __attribute__((ext_vector_type(4))) float v4f;
typedef __attribute__((ext_vector_type(4))) int   v4i;

#define AS1 __attribute__((address_space(1)))
#define AS3 __attribute__((address_space(3)))

// ---- CDNA5 async global<->LDS copy helpers (ASYNCcnt path) ----------------
// Builtins confirmed present on this toolchain (round-1 diagnostic showed the
// expected param type: int4* in the target address space).

__device__ __forceinline__ void async_load_b128(const void* gsrc, void* lds) {
#if defined(__HIP_DEVICE_COMPILE__)
#if __has_builtin(__builtin_amdgcn_global_load_async_to_lds_b128)
    __builtin_amdgcn_global_load_async_to_lds_b128(
        (AS1 v4i*)gsrc, (AS3 v4i*)lds, 0, 0);
#else
    unsigned           l = (unsigned)(size_t)(AS3 void*)lds;
    unsigned long long g = (unsigned long long)(size_t)gsrc;
    asm volatile("global_load_async_to_lds_b128 %0, %1, off"
                 :: "v"(l), "v"(g) : "memory");
#endif
#else
    (void)gsrc; (void)lds;
#endif
}

__device__ __forceinline__ void async_store_b128(void* gdst, const void* lds) {
#if defined(__HIP_DEVICE_COMPILE__)
#if __has_builtin(__builtin_amdgcn_global_store_async_from_lds_b128)
    __builtin_amdgcn_global_store_async_from_lds_b128(
        (AS1 v4i*)gdst, (AS3 v4i*)lds, 0, 0);
#else
    unsigned           l = (unsigned)(size_t)(AS3 void*)lds;
    unsigned long long g = (unsigned long long)(size_t)gdst;
    asm volatile("global_store_async_from_lds_b128 %0, %1, off"
                 :: "v"(g), "v"(l) : "memory");
#endif
#else
    (void)gdst; (void)lds;
#endif
}

__device__ __forceinline__ void wait_asynccnt0() {
#if defined(__HIP_DEVICE_COMPILE__)
#if __has_builtin(__builtin_amdgcn_s_wait_asynccnt)
    __builtin_amdgcn_s_wait_asynccnt(0);
#else
    asm volatile("s_wait_asynccnt 0" ::: "memory");
#endif
#endif
}

// ---------------------------------------------------------------------------
// One block = 64 rows. Each row = 16 lanes x b128. Phase 1: fire all async
// gathers (table rows -> LDS staging). Phase 2: s_wait_asynccnt 0, then async
// store LDS -> out (contiguous, coalesced). Numeric (-1) positions never touch
// LDS: direct splat global_store_b128.
// ---------------------------------------------------------------------------
__global__ __launch_bounds__(THREADS)
void TransformerWord2VecEncoder_gather_kernel(
    const float* __restrict__ inputs,     // [BATCH*SEQ]  (float-encoded ids / numeric vals)
    const float* __restrict__ tables,     // [3*VOCAB*DIM]
    const int*   __restrict__ table_ids,  // [SEQ] pattern {0,1,2,-1}
    float*       __restrict__ out)        // [BATCH*SEQ*DIM]
{
    __shared__ float smem[THREADS * 4 * ITERS];   // 16 KB staging

    const int t = threadIdx.x;
    const int c = t & (CHUNKS - 1);               // 16B chunk within row: 0..15
    const int r = t >> 4;                         // row-in-group:         0..15
    const long long blockBase = (long long)blockIdx.x * ROWS_PER_BLOCK;

    int       tid[ITERS];
    float     xv [ITERS];
    long long pos[ITERS];

    // Phase 1: issue all gathers (ASYNCcnt += up to ITERS per lane)
    #pragma unroll
    for (int i = 0; i < ITERS; ++i) {
        pos[i] = blockBase + (long long)i * (THREADS / CHUNKS) + r;
        const int s = (int)(pos[i] % SEQ);
        tid[i] = table_ids[s];
        xv[i]  = inputs[pos[i]];
        if (tid[i] >= 0) {
            const long long row = (long long)tid[i] * VOCAB + (long long)(int)xv[i];
            const float* src = tables + row * DIM + (long long)c * 4;
            async_load_b128(src, &smem[(i * THREADS + t) * 4]);
        }
    }

    // All gathers for this wave's lanes are in LDS after this.
    wait_asynccnt0();

    // Phase 2: stream rows out (coalesced 256B per row)
    #pragma unroll
    for (int i = 0; i < ITERS; ++i) {
        float* dst = out + pos[i] * DIM + (long long)c * 4;
        if (tid[i] >= 0) {
            async_store_b128(dst, &smem[(i * THREADS + t) * 4]);
        } else {
            v4f v;
            v.x = xv[i]; v.y = xv[i]; v.z = xv[i]; v.w = xv[i];
            *(v4f*)dst = v;
        }
    }

    // Ensure async stores have drained before wave retires.
    wait_asynccnt0();
}

extern "C" void kernel_launch(void* const* d_in, const int* in_sizes, int n_in,
                              void* d_out, int out_size, void* d_ws, size_t ws_size,
                              hipStream_t stream) {
    (void)in_sizes; (void)n_in; (void)out_size; (void)d_ws; (void)ws_size;

    const float* inputs    = (const float*)d_in[0];   // (256, 512) f32
    const float* tables    = (const float*)d_in[1];   // (3, 100000, 64) f32
    const int*   table_ids = (const int*)  d_in[2];   // (512,) i32
    float*       out       = (float*)d_out;           // (256, 512, 64) f32

    const int grid = NPOS / ROWS_PER_BLOCK;           // 131072 / 64 = 2048 blocks
    hipLaunchKernelGGL(TransformerWord2VecEncoder_gather_kernel,
                       dim3(grid), dim3(THREADS), 0, stream,
                       inputs, tables, table_ids, out);
}